// TritonFFTThor_53584011985635
// MI455X (gfx1250) — compile-verified
//
#include <hip/hip_runtime.h>
#include <math.h>

// ---------------------------------------------------------------------------
// Batched 4096-point complex FFT (8192 rows), fp32.
// One workgroup (256 threads = 8 waves) per row; fully LDS-resident.
// 4096 = 16*16*16 -> three radix-16 stages; each stage is a complex
// 16x16 (DFT matrix) x 16x256 (data) product on the fp32 matrix pipe
// via V_WMMA_F32_16X16X4_F32 (4 chained K=4 WMMAs per 16x16x16 product).
// Input row is DMA'd into padded LDS by the Tensor Data Mover (TDM, pad
// fields insert 1 DWORD per 16 DWORDs); output row is DMA'd back to HBM
// with TENSOR_STORE_FROM_LDS from an unpadded staging buffer.
// Twiddles: 2 sincos + geometric recurrence per tile (trans-op diet).
// ---------------------------------------------------------------------------

typedef float v2f  __attribute__((ext_vector_type(2)));
typedef float v8f  __attribute__((ext_vector_type(8)));
typedef unsigned int u32x4 __attribute__((ext_vector_type(4)));
typedef int   i32x4 __attribute__((ext_vector_type(4)));
typedef int   i32x8 __attribute__((ext_vector_type(8)));

#define N_FFT 4096
// pad 1 float per 16 to kill LDS bank conflicts for stride-256/16/1 gathers
#define LDSPAD(i) ((i) + ((i) >> 4))
#define LDS_ELEMS (N_FFT + (N_FFT >> 4))   // 4352

#define TWO_PI_F 6.283185307179586476925f

__device__ __forceinline__ unsigned ldsByteAddr(const void* p) {
    return (unsigned)(uintptr_t)p;         // low 32 bits of LDS aperture = offset
}

// Build TDM descriptor groups for one contiguous 4096-float row.
// Descriptor bitfields per CDNA5 ISA §8.3 (group0) / §8.4 (group1).
__device__ __forceinline__ void tdm_make_row_desc(const void* gaddr, unsigned lds_addr,
                                                  bool pad, u32x4& g0, i32x8& g1) {
    unsigned long long ga = (unsigned long long)(uintptr_t)gaddr;
    g0[0] = 1u;                                          // count=1 (valid user D#)
    g0[1] = lds_addr;                                    // lds_addr (bytes)
    g0[2] = (unsigned)(ga & 0xFFFFFFFFull);              // global_addr[31:0]
    g0[3] = (unsigned)((ga >> 32) & 0x01FFFFFFull)       // global_addr[56:32]
          | 0x80000000u;                                 // type=2 ("image")
    g1[0] = (2 << 16)                                    // data_size: 4 bytes
          | (pad ? ((1 << 20) | (3 << 22)) : 0);         // pad_enable, interval=16 DW
                                                         // pad_amount=0 -> 1 DWORD
    g1[1] = (int)(4096u << 16);                          // tensor_dim0 = 4096 (lo16)
    g1[2] = (int)(1u << 16);                             // dim0 hi=0 | tensor_dim1 = 1
    g1[3] = (int)(4096u << 16);                          // dim1 hi=0 | tile_dim0 = 4096
    g1[4] = 1;                                           // tile_dim1 = 1, tile_dim2 = 0
    g1[5] = 4096;                                        // tensor_dim0_stride (lo32)
    g1[6] = 0;                                           // stride hi | dim1_stride lo
    g1[7] = 0;
}

// TDM load: global row -> LDS, inserting 1 pad DWORD after every 16 DWORDs.
__device__ __forceinline__ void tdm_load_row_padded(const float* __restrict__ gsrc,
                                                    unsigned lds_addr) {
    u32x4 g0; i32x8 g1;
    tdm_make_row_desc(gsrc, lds_addr, /*pad=*/true, g0, g1);
    i32x4 z4 = {0, 0, 0, 0};
    i32x8 z8 = {0, 0, 0, 0, 0, 0, 0, 0};
    __builtin_amdgcn_tensor_load_to_lds(g0, g1, z4, z4, z8, 0);
}

// TDM store: unpadded LDS staging row -> contiguous global row.
__device__ __forceinline__ void tdm_store_row(float* __restrict__ gdst,
                                              unsigned lds_addr) {
    u32x4 g0; i32x8 g1;
    tdm_make_row_desc(gdst, lds_addr, /*pad=*/false, g0, g1);
    i32x4 z4 = {0, 0, 0, 0};
    i32x8 z8 = {0, 0, 0, 0, 0, 0, 0, 0};
    __builtin_amdgcn_tensor_store_from_lds(g0, g1, z4, z4, z8, 0);
}

// One complex 16x16x16 tile: D = W16 * X (complex). W16 held as per-lane
// A-fragments; X gathered from LDS via precomputed padded indices.
// f32 WMMA has no A/B negate (NEG is CNeg-only) so -Wim is materialized.
__device__ __forceinline__ void cplx_dft16_tile(
    const float (&wfr)[4][2], const float (&wfi)[4][2],
    const float* __restrict__ sre, const float* __restrict__ sim,
    const int (&idx)[4][2],
    v8f& dre, v8f& dim)
{
    v8f accr = {0.f,0.f,0.f,0.f,0.f,0.f,0.f,0.f};
    v8f acci = {0.f,0.f,0.f,0.f,0.f,0.f,0.f,0.f};
#pragma unroll
    for (int s = 0; s < 4; ++s) {
        v2f ar;  ar.x  =  wfr[s][0]; ar.y  =  wfr[s][1];
        v2f ai;  ai.x  =  wfi[s][0]; ai.y  =  wfi[s][1];
        v2f ain; ain.x = -wfi[s][0]; ain.y = -wfi[s][1];
        v2f br;  br.x  = sre[idx[s][0]]; br.y = sre[idx[s][1]];
        v2f bi;  bi.x  = sim[idx[s][0]]; bi.y = sim[idx[s][1]];
        // Dre += Wre*Xre - Wim*Xim ; Dim += Wre*Xim + Wim*Xre
        accr = __builtin_amdgcn_wmma_f32_16x16x4_f32(false, ar,  false, br, (short)0, accr, false, false);
        accr = __builtin_amdgcn_wmma_f32_16x16x4_f32(false, ain, false, bi, (short)0, accr, false, false);
        acci = __builtin_amdgcn_wmma_f32_16x16x4_f32(false, ar,  false, bi, (short)0, acci, false, false);
        acci = __builtin_amdgcn_wmma_f32_16x16x4_f32(false, ai,  false, br, (short)0, acci, false, false);
    }
    dre = accr;
    dim = acci;
}

__global__ __launch_bounds__(256)
void fft4096_wmma_kernel(const float* __restrict__ xre,
                         const float* __restrict__ xim,
                         float* __restrict__ yre,
                         float* __restrict__ yim)
{
    __shared__ float Are[LDS_ELEMS], Aim[LDS_ELEMS];
    __shared__ float Bre[LDS_ELEMS], Bim[LDS_ELEMS];

    const int tid  = threadIdx.x;
    const int lane = tid & 31;
    const int wv   = tid >> 5;       // wave id 0..7
    const int nl   = lane & 15;      // tile column / A-row
    const int half = lane >> 4;      // selects K pair (A/B) and C row offset

    const size_t row = blockIdx.x;
    const float* gre = xre + row * (size_t)N_FFT;
    const float* gim = xim + row * (size_t)N_FFT;

    // ---- TDM: async-DMA the row into padded LDS (wave 0 drives both) ----
    if (wv == 0) {
        tdm_load_row_padded(gre, ldsByteAddr(&Are[0]));
        tdm_load_row_padded(gim, ldsByteAddr(&Aim[0]));
        __builtin_amdgcn_s_wait_tensorcnt(0);
    }

    // ---- per-lane W16 A-fragments (ISA A layout: M=lane&15, K=4s+2*half+v)
    // (computed while the TDM fills LDS)
    float wfr[4][2], wfi[4][2];
#pragma unroll
    for (int s = 0; s < 4; ++s) {
#pragma unroll
        for (int v = 0; v < 2; ++v) {
            int k = 4*s + 2*half + v;
            float ang = -(TWO_PI_F / 16.0f) * (float)((nl * k) & 15);
            float sv, cv; __sincosf(ang, &sv, &cv);
            wfr[s][v] = cv; wfi[s][v] = sv;
        }
    }

    // ---- Stage-B twiddle factors W256^{m*nl}, m=8*half+r: tile-independent,
    // so build once via 2 sincos + geometric recurrence.
    float twBc[8], twBs[8];
    {
        float ss, cs;   // step exp(-i*2pi*nl/256)
        __sincosf(-(TWO_PI_F / 256.0f) * (float)nl, &ss, &cs);
        float sb, cb;   // base exp(-i*2pi*(8*half)*nl/256)
        __sincosf(-(TWO_PI_F / 256.0f) * (float)((8 * half * nl) & 255), &sb, &cb);
#pragma unroll
        for (int r = 0; r < 8; ++r) {
            twBc[r] = cb; twBs[r] = sb;
            float nc = cb * cs - sb * ss;
            float ns = cb * ss + sb * cs;
            cb = nc; sb = ns;
        }
    }

    __syncthreads();

    // ================= Stage A: DFT-16 over stride-256 axis =================
    // t[k1][n2] = sum_n1 W16[k1][n1] * x[n1*256 + n2];  twiddle W4096^{k1*n2}
#pragma unroll
    for (int tt = 0; tt < 2; ++tt) {
        const int t = wv + 8 * tt;
        const int c = t * 16 + nl;                    // n2
        int idx[4][2];
#pragma unroll
        for (int s = 0; s < 4; ++s)
#pragma unroll
            for (int v = 0; v < 2; ++v) {
                int k = 4*s + 2*half + v;             // n1
                idx[s][v] = LDSPAD(k * 256 + c);
            }
        v8f dre, dim;
        cplx_dft16_tile(wfr, wfi, Are, Aim, idx, dre, dim);

        // twiddle tw(m) = exp(-i*2pi*c*m/4096), m = 8*half + r: recurrence
        float ss, cs;
        __sincosf(-(TWO_PI_F / 4096.0f) * (float)c, &ss, &cs);
        float sb, cb;
        __sincosf(-(TWO_PI_F / 4096.0f) * (float)(((8 * half) * c) & 4095), &sb, &cb);
#pragma unroll
        for (int r = 0; r < 8; ++r) {
            int m = r + 8 * half;                     // k1
            float re = dre[r] * cb - dim[r] * sb;
            float im = dre[r] * sb + dim[r] * cb;
            int o = LDSPAD(m * 256 + c);
            Bre[o] = re; Bim[o] = im;
            float nc = cb * cs - sb * ss;             // rotate by step
            float ns = cb * ss + sb * cs;
            cb = nc; sb = ns;
        }
    }
    __syncthreads();

    // ================= Stage B: DFT-16 over stride-16 axis ==================
    // s[j1][k1][m2] = sum_m1 W16[j1][m1] * t[k1][m1*16+m2]; twiddle W256^{j1*m2}
#pragma unroll
    for (int tt = 0; tt < 2; ++tt) {
        const int t = wv + 8 * tt;                    // k1
        int idx[4][2];
#pragma unroll
        for (int s = 0; s < 4; ++s)
#pragma unroll
            for (int v = 0; v < 2; ++v) {
                int k = 4*s + 2*half + v;             // m1
                idx[s][v] = LDSPAD(t * 256 + k * 16 + nl);
            }
        v8f dre, dim;
        cplx_dft16_tile(wfr, wfi, Bre, Bim, idx, dre, dim);
#pragma unroll
        for (int r = 0; r < 8; ++r) {
            int m = r + 8 * half;                     // j1
            float re = dre[r] * twBc[r] - dim[r] * twBs[r];
            float im = dre[r] * twBs[r] + dim[r] * twBc[r];
            int o = LDSPAD(m * 256 + t * 16 + nl);    // store s[j1][k1][m2]
            Are[o] = re; Aim[o] = im;
        }
    }
    __syncthreads();

    // ================= Stage C: DFT-16 over unit-stride axis ================
    // w[j2][j1][k1] = sum_m2 W16[j2][m2] * s[j1][k1][m2]
    // Y[k1 + 16*j1 + 256*j2] = w[j2][j1][k1]
    // Results staged unpadded into the free B buffers (exact output order),
    // then DMA'd to HBM with TENSOR_STORE_FROM_LDS.
#pragma unroll
    for (int tt = 0; tt < 2; ++tt) {
        const int t = wv + 8 * tt;                    // j1
        int idx[4][2];
#pragma unroll
        for (int s = 0; s < 4; ++s)
#pragma unroll
            for (int v = 0; v < 2; ++v) {
                int k = 4*s + 2*half + v;             // m2
                idx[s][v] = LDSPAD(t * 256 + nl * 16 + k);
            }
        v8f dre, dim;
        cplx_dft16_tile(wfr, wfi, Are, Aim, idx, dre, dim);
#pragma unroll
        for (int r = 0; r < 8; ++r) {
            int m = r + 8 * half;                     // j2
            int o = m * 256 + t * 16 + nl;            // 256*j2 + 16*j1 + k1
            Bre[o] = dre[r];
            Bim[o] = dim[r];
        }
    }
    __syncthreads();

    if (wv == 0) {
        float* outre = yre + row * (size_t)N_FFT;
        float* outim = yim + row * (size_t)N_FFT;
        tdm_store_row(outre, ldsByteAddr(&Bre[0]));
        tdm_store_row(outim, ldsByteAddr(&Bim[0]));
        __builtin_amdgcn_s_wait_tensorcnt(0);
    }
}

extern "C" void kernel_launch(void* const* d_in, const int* in_sizes, int n_in,
                              void* d_out, int out_size, void* d_ws, size_t ws_size,
                              hipStream_t stream) {
    const float* xre = (const float*)d_in[0];
    const float* xim = (const float*)d_in[1];
    float* out = (float*)d_out;

    const int total = in_sizes[0];          // B * N
    const int rows  = total / N_FFT;        // 8192
    float* yre = out;
    float* yim = out + (size_t)total;       // outputs concatenated: re then im

    fft4096_wmma_kernel<<<rows, 256, 0, stream>>>(xre, xim, yre, yim);
}